// SPAttention_33079838114044
// MI455X (gfx1250) — compile-verified
//
#include <hip/hip_runtime.h>
#include <math.h>

// CDNA5 / gfx1250: wave32, WMMA f32 16x16x4 (A/B = 2 VGPRs, C/D = 8 VGPRs)
typedef float v2f __attribute__((ext_vector_type(2)));
typedef float v8f __attribute__((ext_vector_type(8)));

#define SCALE 0.08838834764831845f   // (1024/8)^-0.5

// ---------------------------------------------------------------------------
// Kernel 1: qk[64][2048] = xf[64][1024] @ W15^T   (NT gemm, fp32 WMMA)
// one wave per 16x16 C tile; grid = (2048/16, 64/16)
// A layout (16x4 f32):  lane L holds A[L&15][2*(L>>4) + {0,1}] in {a.x,a.y}
// B layout (4x16 f32):  lane L holds B[2*(L>>4) + {0,1}][L&15] = W[n][k...] (NT)
// ---------------------------------------------------------------------------
__global__ __launch_bounds__(32) void qk_gemm(const float* __restrict__ xf,
                                              const float* __restrict__ W15,
                                              float* __restrict__ qk) {
  const int lane = threadIdx.x;
  const int half = lane >> 4, lm = lane & 15;
  const int ntile = blockIdx.x, mtile = blockIdx.y;

  const float* __restrict__ arow = xf  + (size_t)(mtile * 16 + lm) * 1024 + 2 * half;
  const float* __restrict__ brow = W15 + (size_t)(ntile * 16 + lm) * 1024 + 2 * half;

  v8f c = {};
#pragma unroll 8
  for (int k = 0; k < 1024; k += 4) {
    v2f a = *(const v2f*)(arow + k);
    v2f b = *(const v2f*)(brow + k);
    c = __builtin_amdgcn_wmma_f32_16x16x4_f32(false, a, false, b, (short)0, c,
                                              false, false);
  }
  // C layout: VGPR r -> (M = r + 8*half, N = lm)
  const int m0 = mtile * 16 + 8 * half;
  const int n  = ntile * 16 + lm;
#pragma unroll
  for (int r = 0; r < 8; ++r) qk[(size_t)(m0 + r) * 2048 + n] = c[r];
}

// ---------------------------------------------------------------------------
// Kernel 2: per (b,h): attn = SCALE * q @ k^T (16x16), custom exp-norm -> a_ws
// ---------------------------------------------------------------------------
__global__ __launch_bounds__(256) void attn_kernel(const float* __restrict__ qk,
                                                   float* __restrict__ a_ws) {
  __shared__ float qs[16 * 128];
  __shared__ float ks[16 * 128];
  __shared__ float ss[16 * 16];
  const int tid = threadIdx.x;
  const int bh = blockIdx.x, b = bh >> 3, h = bh & 7;

#pragma unroll
  for (int i = 0; i < 8; ++i) {
    int idx = tid * 8 + i;            // 0..2047
    int row = idx >> 7, d = idx & 127;
    qs[idx] = qk[(size_t)(b * 16 + row) * 2048 + h * 128 + d];
    ks[idx] = qk[(size_t)(b * 16 + row) * 2048 + 1024 + h * 128 + d];
  }
  __syncthreads();

  const int i = tid >> 4, j = tid & 15;
  float s = 0.f;
#pragma unroll 4
  for (int d = 0; d < 128; ++d) s += qs[i * 128 + d] * ks[j * 128 + d];
  s *= SCALE;
  ss[i * 16 + j] = s;
  __syncthreads();

  float amax = -3.402823466e38f, denom = 0.f;
#pragma unroll
  for (int t = 0; t < 16; ++t) {
    float v = ss[i * 16 + t];
    amax = fmaxf(amax, v);
    denom += __expf(v);
  }
  const float mask = ((s > 0.5f) || (s == amax)) ? 1.0f : 0.0f;
  a_ws[bh * 256 + i * 16 + j] = __expf(s * mask) / denom;
}

// ---------------------------------------------------------------------------
// Kernel 3: per (b,h,nn):  Y(16x64) = a(16x16) @ Xperm(16x64)
//                          Z(16x64) = Y @ proj_w^T + proj_b  -> interleaved out
// xo_final[b,h,q,:] = x_o_flat[b, rot_right1_7b(h*16+q)*8192 : +8192]
//   rot_right1(h*16+q) = h*8 + (q>>1) + ((q&1)<<6)
// out[b, 2h+(p>>3), (p&7)*128+nn, c]
// grid = (32, 16), block = 256 (8 waves, wave w -> nn = 8*blockIdx.y + w)
// ---------------------------------------------------------------------------
__global__ __launch_bounds__(256) void av_proj(const float* __restrict__ xo,
                                               const float* __restrict__ a_ws,
                                               const float* __restrict__ projw,
                                               const float* __restrict__ projb,
                                               float* __restrict__ out) {
  __shared__ float a_lds[16 * 20];          // pitch 20: conflict-free A reads
  __shared__ float wp_lds[64 * 68];         // pitch 68: conflict-free B reads
  __shared__ float y_lds[8 * 16 * 68];      // per-wave Y strip, pitch 68

  const int tid = threadIdx.x;
  const int bh = blockIdx.x, b = bh >> 3, h = bh & 7;
  const int wv = tid >> 5, lane = tid & 31, half = lane >> 4, lm = lane & 15;
  const int nn = blockIdx.y * 8 + wv;

  a_lds[(tid >> 4) * 20 + (tid & 15)] = a_ws[bh * 256 + tid];
#pragma unroll
  for (int t = 0; t < 16; ++t) {
    int idx = tid + t * 256;                // 0..4095
    wp_lds[(idx >> 6) * 68 + (idx & 63)] = projw[idx];
  }
  __syncthreads();

  const float* __restrict__ xob = xo + (size_t)b * 1048576 + (size_t)nn * 64;
  float* __restrict__ yb = y_lds + wv * 16 * 68;

  // ---- GEMM1: Y = a @ X, K = 16 (4 wmma per 16-col tile) ----
#pragma unroll
  for (int t = 0; t < 4; ++t) {
    v8f y = {};
#pragma unroll
    for (int kk = 0; kk < 4; ++kk) {
      const int k0 = 4 * kk + 2 * half;             // even
      v2f a = *(const v2f*)(a_lds + lm * 20 + k0);  // A[lm][k0], A[lm][k0+1]
      const int q0 = k0, q1 = k0 + 1;
      const int c0 = h * 8 + (q0 >> 1) + ((q0 & 1) << 6);   // rot_right1(h*16+q)
      const int c1 = h * 8 + (q1 >> 1) + ((q1 & 1) << 6);
      v2f bb;
      bb.x = xob[(size_t)c0 * 8192 + t * 16 + lm];
      bb.y = xob[(size_t)c1 * 8192 + t * 16 + lm];
      y = __builtin_amdgcn_wmma_f32_16x16x4_f32(false, a, false, bb, (short)0, y,
                                                false, false);
    }
#pragma unroll
    for (int r = 0; r < 8; ++r) yb[(r + 8 * half) * 68 + t * 16 + lm] = y[r];
  }
  __syncthreads();

  // ---- GEMM2: Z = Y @ proj_w^T, K = 64 (16 wmma per 16-col tile) ----
#pragma unroll
  for (int t2 = 0; t2 < 4; ++t2) {
    v8f z = {};
#pragma unroll
    for (int kk = 0; kk < 16; ++kk) {
      const int k0 = 4 * kk + 2 * half;                         // even
      v2f a = *(const v2f*)(yb + lm * 68 + k0);                 // Y[lm][k0,k0+1]
      v2f bb = *(const v2f*)(wp_lds + (t2 * 16 + lm) * 68 + k0);// Wp[n][k0,k0+1] (NT)
      z = __builtin_amdgcn_wmma_f32_16x16x4_f32(false, a, false, bb, (short)0, z,
                                                false, false);
    }
    const int co = t2 * 16 + lm;
    const float pb = projb[co];
#pragma unroll
    for (int r = 0; r < 8; ++r) {
      const int p = r + 8 * half;
      const int P = 2 * h + (p >> 3);
      const int n = (p & 7) * 128 + nn;
      out[(size_t)b * 1048576 + (size_t)P * 65536 + (size_t)n * 64 + co] =
          z[r] + pb;
    }
  }
}

// ---------------------------------------------------------------------------
extern "C" void kernel_launch(void* const* d_in, const int* in_sizes, int n_in,
                              void* d_out, int out_size, void* d_ws, size_t ws_size,
                              hipStream_t stream) {
  const float* x      = (const float*)d_in[0];   // [4,16,32,32] == xf[64][1024]
  const float* x_o    = (const float*)d_in[1];   // [4,16,1024,64]
  const float* W_qk   = (const float*)d_in[2];   // [16,2048,1024]
  const float* proj_w = (const float*)d_in[3];   // [64,64]
  const float* proj_b = (const float*)d_in[4];   // [64]
  // d_in[5], d_in[6]: H, W scalars (unused; geometry is fixed)

  const float* W15 = W_qk + (size_t)15 * 2048 * 1024;  // only i=15 reaches out

  float* qk_ws = (float*)d_ws;                   // 64*2048 floats
  float* a_ws  = qk_ws + 64 * 2048;              // 32*256 floats
  float* out   = (float*)d_out;                  // [4,16,1024,64]

  qk_gemm<<<dim3(128, 4), 32, 0, stream>>>(x, W15, qk_ws);
  attn_kernel<<<dim3(32), 256, 0, stream>>>(qk_ws, a_ws);
  av_proj<<<dim3(32, 16), 256, 0, stream>>>(x_o, a_ws, proj_w, proj_b, out);
}